// PPIModel_80401787781821
// MI455X (gfx1250) — compile-verified
//
#include <hip/hip_runtime.h>
#include <hip/hip_bf16.h>
#include <math.h>

// ---------------------------------------------------------------------------
// PPI GCN forward: one workgroup per graph.
//   B=8192 graphs, 286 nodes/graph (3 feats), 4000 edges/graph.
//   Stage edges+features in LDS (async global->LDS), degrees + 2x SpMM via
//   ds_add_f32 atomics, 3x3 transforms via V_WMMA_F32_16X16X4_F32,
//   858-dot FC + sigmoid.
// Memory-bound: ~290 MB of HBM traffic -> ~13us floor at 23.3 TB/s.
// ---------------------------------------------------------------------------

#define NPG   286
#define EPG   4000
#define FDIM  3
#define TPB   256
#define NTILES ((NPG + 15) / 16)   /* 18 */

typedef float v2f __attribute__((ext_vector_type(2)));
typedef float v8f __attribute__((ext_vector_type(8)));

// exact pointer types for the async global->LDS builtins (from hipcc sema):
//   b128: int __attribute__((vector_size(16))) __device__ *   (as1 src)
//   b64 : int __attribute__((vector_size(8)))  __device__ *
typedef int v4i_vs __attribute__((vector_size(16)));
typedef int v2i_vs __attribute__((vector_size(8)));
#define AS1 __attribute__((address_space(1)))
#define AS3 __attribute__((address_space(3)))

#if defined(__AMDGCN__) && \
    __has_builtin(__builtin_amdgcn_global_load_async_to_lds_b128) && \
    __has_builtin(__builtin_amdgcn_global_load_async_to_lds_b64)
#define HAVE_ASYNC_LDS 1
#endif

__device__ __forceinline__ void async_wait_all() {
#if defined(HAVE_ASYNC_LDS)
#if __has_builtin(__builtin_amdgcn_s_wait_asynccnt)
    __builtin_amdgcn_s_wait_asynccnt(0);
#else
    asm volatile("s_wait_asynccnt 0x0" ::: "memory");
#endif
#endif
}

__global__ __launch_bounds__(TPB) void ppi_gcn_kernel(
    const float* __restrict__ feat,
    const int*   __restrict__ src,
    const int*   __restrict__ dst,
    const float* __restrict__ W1, const float* __restrict__ b1,
    const float* __restrict__ W2, const float* __restrict__ b2,
    const float* __restrict__ Wfc, const float* __restrict__ bfc,
    float* __restrict__ out)
{
    __shared__ __align__(16) unsigned sSrc[EPG];       // 16000 B (local ids after pass 1)
    __shared__ __align__(16) unsigned sDst[EPG];       // 16000 B
    __shared__ __align__(16) float    sX  [NPG * FDIM];// current layer input (pre-scaled)
    __shared__ __align__(16) float    sAgg[NPG * FDIM];// scatter accumulator
    __shared__ float sNO[NPG];                          // deg_out -> norm_out
    __shared__ float sNI[NPG];                          // deg_in  -> norm_in
    __shared__ float sRed;

    const int g    = blockIdx.x;
    const int tid  = threadIdx.x;
    const unsigned base = (unsigned)g * NPG;

    const unsigned* gs = (const unsigned*)src + (size_t)g * EPG;
    const unsigned* gd = (const unsigned*)dst + (size_t)g * EPG;
    const float*    gx = feat + (size_t)g * (NPG * FDIM);

    // ---- stage edge lists + node features into LDS -------------------------
#if defined(HAVE_ASYNC_LDS)
    // edges: 4000 u32 each = 1000 b128 transfers per list (16B aligned)
    for (int it = 0; it < (EPG / 4 + TPB - 1) / TPB; ++it) {
        int idx = tid + it * TPB;
        if (idx < EPG / 4) {
            __builtin_amdgcn_global_load_async_to_lds_b128(
                (AS1 v4i_vs*)(gs + idx * 4), (AS3 v4i_vs*)(&sSrc[idx * 4]), 0, 0);
            __builtin_amdgcn_global_load_async_to_lds_b128(
                (AS1 v4i_vs*)(gd + idx * 4), (AS3 v4i_vs*)(&sDst[idx * 4]), 0, 0);
        }
    }
    // features: 858 f32 = 429 b64 transfers (gx is only 8B-aligned for odd g)
    for (int it = 0; it < (NPG * FDIM / 2 + TPB - 1) / TPB; ++it) {
        int idx = tid + it * TPB;
        if (idx < NPG * FDIM / 2) {
            __builtin_amdgcn_global_load_async_to_lds_b64(
                (AS1 v2i_vs*)(gx + idx * 2), (AS3 v2i_vs*)(&sX[idx * 2]), 0, 0);
        }
    }
#else
    for (int e = tid; e < EPG; e += TPB) { sSrc[e] = gs[e]; sDst[e] = gd[e]; }
    for (int i = tid; i < NPG * FDIM; i += TPB) sX[i] = gx[i];
#endif
    // zero degree accumulators while the async DMA is in flight
    for (int n = tid; n < NPG; n += TPB) { sNO[n] = 0.0f; sNI[n] = 0.0f; }
    if (tid == 0) sRed = 0.0f;
    async_wait_all();
    __syncthreads();

    // ---- degrees (convert ids to graph-local in place) ---------------------
    for (int e = tid; e < EPG; e += TPB) {
        unsigned s = sSrc[e] - base; sSrc[e] = s;
        unsigned d = sDst[e] - base; sDst[e] = d;
        atomicAdd(&sNO[s], 1.0f);
        atomicAdd(&sNI[d], 1.0f);
    }
    __syncthreads();

    // ---- norms; pre-scale features by norm_out -----------------------------
    for (int n = tid; n < NPG; n += TPB) {
        float no = rsqrtf(fmaxf(sNO[n], 1.0f));
        float ni = rsqrtf(fmaxf(sNI[n], 1.0f));
        sNO[n] = no; sNI[n] = ni;
        sX[n * 3 + 0] *= no;
        sX[n * 3 + 1] *= no;
        sX[n * 3 + 2] *= no;
    }
    __syncthreads();

    const int lane = tid & 31;
    const int wid  = tid >> 5;
    const int mrow = lane & 15;            // M (A) / N (B,D) position
    const int khi  = (lane < 16) ? 0 : 2;  // K base held by this lane half

    // ---- two GraphConv layers ----------------------------------------------
    for (int layer = 0; layer < 2; ++layer) {
        const float* W    = layer ? W2 : W1;
        const float* bias = layer ? b2 : b1;

        for (int i = tid; i < NPG * FDIM; i += TPB) sAgg[i] = 0.0f;
        __syncthreads();

        // SpMM scatter: agg[dst] += x_scaled[src]
        for (int e = tid; e < EPG; e += TPB) {
            unsigned s = sSrc[e], d = sDst[e];
            float x0 = sX[s * 3 + 0];
            float x1 = sX[s * 3 + 1];
            float x2 = sX[s * 3 + 2];
            atomicAdd(&sAgg[d * 3 + 0], x0);
            atomicAdd(&sAgg[d * 3 + 1], x1);
            atomicAdd(&sAgg[d * 3 + 2], x2);
        }
        __syncthreads();

        // B operand: 3x3 W padded to 4x16 (K rows split across lane halves,
        // mirroring the documented 16x4 fp32 A layout).
        v2f bm;
        bm.x = (mrow < 3) ? W[khi * 3 + mrow] : 0.0f;                  // K = khi
        bm.y = (mrow < 3 && (khi + 1) < 3) ? W[(khi + 1) * 3 + mrow]   // K = khi+1
                                           : 0.0f;
        const float bcol = (mrow < 3) ? bias[mrow] : 0.0f;

        // D = (agg * norm_in) @ W for 16-node tiles, 8 waves cover 18 tiles
        for (int t = wid; t < NTILES; t += TPB / 32) {
            int node = t * 16 + mrow;
            v2f a; a.x = 0.0f; a.y = 0.0f;
            if (node < NPG) {
                float ni = sNI[node];
                a.x = sAgg[node * 3 + khi] * ni;                 // K = 0 or 2
                if (khi == 0) a.y = sAgg[node * 3 + 1] * ni;     // K = 1 (K=3 pad)
            }
            v8f c = {0.0f, 0.0f, 0.0f, 0.0f, 0.0f, 0.0f, 0.0f, 0.0f};
            v8f acc = __builtin_amdgcn_wmma_f32_16x16x4_f32(
                false, a, false, bm, (short)0, c, false, false);

            if (mrow < 3) {   // only output columns 0..2 are real
#pragma unroll
                for (int r = 0; r < 8; ++r) {
                    int M  = r + ((lane < 16) ? 0 : 8);
                    int nd = t * 16 + M;
                    if (nd < NPG) {
                        float h = fmaxf(acc[r] + bcol, 0.0f);    // relu(agg@W + b)
                        if (layer == 0) h *= sNO[nd];            // pre-scale for layer 2
                        sX[nd * 3 + mrow] = h;
                    }
                }
            }
        }
        __syncthreads();
    }

    // ---- FC: sigmoid(x_flat . Wfc + bfc) -----------------------------------
    float part = 0.0f;
    for (int i = tid; i < NPG * FDIM; i += TPB) part += sX[i] * Wfc[i];
    atomicAdd(&sRed, part);
    __syncthreads();
    if (tid == 0) {
        float v = sRed + bfc[0];
        out[g] = 1.0f / (1.0f + expf(-v));
    }
}

extern "C" void kernel_launch(void* const* d_in, const int* in_sizes, int n_in,
                              void* d_out, int out_size, void* d_ws, size_t ws_size,
                              hipStream_t stream) {
    (void)in_sizes; (void)n_in; (void)d_ws; (void)ws_size;
    const float* feat = (const float*)d_in[0];
    const int*   src  = (const int*)  d_in[1];
    const int*   dst  = (const int*)  d_in[2];
    const float* W1   = (const float*)d_in[3];
    const float* b1   = (const float*)d_in[4];
    const float* W2   = (const float*)d_in[5];
    const float* b2   = (const float*)d_in[6];
    const float* Wfc  = (const float*)d_in[7];
    const float* bfc  = (const float*)d_in[8];
    float* out = (float*)d_out;

    const int nGraphs = out_size;  // 8192, one output per graph
    hipLaunchKernelGGL(ppi_gcn_kernel, dim3(nGraphs), dim3(TPB), 0, stream,
                       feat, src, dst, W1, b1, W2, b2, Wfc, bfc, out);
}